// SimpleNetDeep_31516470018049
// MI455X (gfx1250) — compile-verified
//
#include <hip/hip_runtime.h>
#include <hip/hip_bf16.h>

#define N_NODES 100000
#define N_EDGES 1600000

typedef __attribute__((ext_vector_type(2))) float v2f;
typedef __attribute__((ext_vector_type(8))) float v8f;

static inline int cdiv_ll(long long a, long long b) { return (int)((a + b - 1) / b); }

// ---------------- degree / normalization ----------------

__global__ void k_init_deg(float* __restrict__ deg) {
    int i = blockIdx.x * blockDim.x + threadIdx.x;
    if (i < N_NODES) deg[i] = 1.0f;  // self-loop contributes 1 to every node's degree
}

__global__ void k_count_deg(const int* __restrict__ dst, float* __restrict__ deg) {
    int e = blockIdx.x * blockDim.x + threadIdx.x;
    if (e < N_EDGES) unsafeAtomicAdd(&deg[dst[e]], 1.0f);
}

__global__ void k_dinv(const float* __restrict__ deg, float* __restrict__ dinv) {
    int i = blockIdx.x * blockDim.x + threadIdx.x;
    if (i < N_NODES) dinv[i] = rsqrtf(deg[i]);  // deg >= 1 always (self-loops)
}

// ---------------- WMMA fp32 GEMM:  H = act(X) @ W ----------------
// One wave -> one 16x16 tile of H. K advances 4 per V_WMMA_F32_16X16X4_F32.
// A (16x4 f32):  lane m=lane&15 holds K = (lane>>4)*2 + {0,1} in 2 VGPRs.
// B (4x16 f32):  lane n=lane&15 holds K = (lane>>4)*2 + {0,1} in 2 VGPRs.
// D (16x16 f32): VGPR i holds row (lane>>4)*8 + i, col lane&15.
template <int FIN, int FOUT, bool RELU>
__global__ void k_gemm_wmma(const float* __restrict__ X, const float* __restrict__ W,
                            float* __restrict__ H) {
    constexpr int COLT = FOUT / 16;
    constexpr int ROWT = (N_NODES + 15) / 16;
    int gtid = blockIdx.x * blockDim.x + threadIdx.x;
    int wave = gtid >> 5;
    if (wave >= ROWT * COLT) return;  // uniform per wave

    int lane = threadIdx.x & 31;
    int rt = wave / COLT;
    int ct = wave % COLT;
    int row0 = rt * 16;
    int m = lane & 15;
    int kb = (lane >> 4) * 2;  // 0 or 2
    int row = row0 + m;
    bool ok = row < N_NODES;                     // per-lane tail mask (selects, no branch)
    const float* xr = X + (size_t)(ok ? row : 0) * FIN;
    int n = ct * 16 + m;

    v8f acc = {0.f, 0.f, 0.f, 0.f, 0.f, 0.f, 0.f, 0.f};

#pragma unroll
    for (int k = 0; k < FIN; k += 4) {
        float a0 = xr[k + kb];
        float a1 = xr[k + kb + 1];
        if (RELU) {
            a0 = fmaxf(a0, 0.0f);
            a1 = fmaxf(a1, 0.0f);
        }
        a0 = ok ? a0 : 0.0f;
        a1 = ok ? a1 : 0.0f;
        v2f a;
        a.x = a0;
        a.y = a1;
        v2f b;
        b.x = W[(size_t)(k + kb) * FOUT + n];
        b.y = W[(size_t)(k + kb + 1) * FOUT + n];
        acc = __builtin_amdgcn_wmma_f32_16x16x4_f32(false, a, false, b, (short)0, acc,
                                                    false, false);
    }

    int mb = (lane >> 4) * 8;
#pragma unroll
    for (int i = 0; i < 8; ++i) {
        int r = row0 + mb + i;
        if (r < N_NODES) H[(size_t)r * FOUT + ct * 16 + m] = acc[i];
    }
}

// ---------------- out = dinv^2 * H + bias  (self-loop term + bias) ----------------
template <int F>
__global__ void k_self_bias(const float* __restrict__ H, const float* __restrict__ dinv,
                            const float* __restrict__ b, float* __restrict__ out) {
    long long t = blockIdx.x * (long long)blockDim.x + threadIdx.x;
    if (t >= (long long)N_NODES * F) return;
    int i = (int)(t / F);
    int f = (int)(t % F);
    float di = dinv[i];
    out[t] = di * di * H[t] + b[f];
}

// ---------------- edge scatter-add:  out[dst] += dinv[src]*dinv[dst] * H[src] ----------------
// F/2 threads per edge, float2 gather, two native f32 atomics per thread.
template <int F>
__global__ void k_scatter(const float* __restrict__ H, const int* __restrict__ src,
                          const int* __restrict__ dst, const float* __restrict__ dinv,
                          float* __restrict__ out) {
    constexpr int TPE = F / 2;
    long long t = blockIdx.x * (long long)blockDim.x + threadIdx.x;
    long long e = t / TPE;
    if (e >= N_EDGES) return;
    int c = (int)(t % TPE);
    int s = src[e];
    int d = dst[e];
    float w = dinv[s] * dinv[d];
    const float2* hp = reinterpret_cast<const float2*>(H + (size_t)s * F);
    float2 v = hp[c];
    float* op = out + (size_t)d * F + 2 * c;
    unsafeAtomicAdd(op + 0, w * v.x);
    unsafeAtomicAdd(op + 1, w * v.y);
}

// ---------------- launch ----------------

extern "C" void kernel_launch(void* const* d_in, const int* in_sizes, int n_in,
                              void* d_out, int out_size, void* d_ws, size_t ws_size,
                              hipStream_t stream) {
    (void)in_sizes; (void)n_in; (void)out_size; (void)ws_size;

    const float* x  = (const float*)d_in[0];
    const int*   ei = (const int*)d_in[1];
    const int* srcp = ei;             // edge_index[0]
    const int* dstp = ei + N_EDGES;   // edge_index[1]
    const float* W1 = (const float*)d_in[2];
    const float* b1 = (const float*)d_in[3];
    const float* W2 = (const float*)d_in[4];
    const float* b2 = (const float*)d_in[5];
    const float* W3 = (const float*)d_in[6];
    const float* b3 = (const float*)d_in[7];
    const float* W4 = (const float*)d_in[8];
    const float* b4 = (const float*)d_in[9];
    float* out = (float*)d_out;

    float* ws   = (float*)d_ws;
    float* deg  = ws;
    float* dinv = deg + N_NODES;
    float* H    = dinv + N_NODES;                  // N x 64
    float* aggA = H + (size_t)N_NODES * 64;        // N x 64
    float* aggB = aggA + (size_t)N_NODES * 64;     // N x 64

    const int B = 256;
    const long long ROWT = (N_NODES + 15) / 16;

    // normalization
    k_init_deg<<<cdiv_ll(N_NODES, B), B, 0, stream>>>(deg);
    k_count_deg<<<cdiv_ll(N_EDGES, B), B, 0, stream>>>(dstp, deg);
    k_dinv<<<cdiv_ll(N_NODES, B), B, 0, stream>>>(deg, dinv);

    // layer 1: x(128) -> 64, no input relu
    k_gemm_wmma<128, 64, false>
        <<<cdiv_ll(ROWT * 4 * 32, B), B, 0, stream>>>(x, W1, H);
    k_self_bias<64><<<cdiv_ll((long long)N_NODES * 64, B), B, 0, stream>>>(H, dinv, b1, aggA);
    k_scatter<64><<<cdiv_ll((long long)N_EDGES * 32, B), B, 0, stream>>>(H, srcp, dstp, dinv, aggA);

    // layer 2: relu(aggA)(64) -> 64
    k_gemm_wmma<64, 64, true>
        <<<cdiv_ll(ROWT * 4 * 32, B), B, 0, stream>>>(aggA, W2, H);
    k_self_bias<64><<<cdiv_ll((long long)N_NODES * 64, B), B, 0, stream>>>(H, dinv, b2, aggB);
    k_scatter<64><<<cdiv_ll((long long)N_EDGES * 32, B), B, 0, stream>>>(H, srcp, dstp, dinv, aggB);

    // layer 3: relu(aggB)(64) -> 64
    k_gemm_wmma<64, 64, true>
        <<<cdiv_ll(ROWT * 4 * 32, B), B, 0, stream>>>(aggB, W3, H);
    k_self_bias<64><<<cdiv_ll((long long)N_NODES * 64, B), B, 0, stream>>>(H, dinv, b3, aggA);
    k_scatter<64><<<cdiv_ll((long long)N_EDGES * 32, B), B, 0, stream>>>(H, srcp, dstp, dinv, aggA);

    // layer 4: relu(aggA)(64) -> 16, output straight to d_out (no relu after)
    k_gemm_wmma<64, 16, true>
        <<<cdiv_ll(ROWT * 1 * 32, B), B, 0, stream>>>(aggA, W4, H);
    k_self_bias<16><<<cdiv_ll((long long)N_NODES * 16, B), B, 0, stream>>>(H, dinv, b4, out);
    k_scatter<16><<<cdiv_ll((long long)N_EDGES * 8, B), B, 0, stream>>>(H, srcp, dstp, dinv, out);
}